// back_proj_net_48627619725578
// MI455X (gfx1250) — compile-verified
//
#include <hip/hip_runtime.h>

#define AS1 __attribute__((address_space(1)))
#define AS3 __attribute__((address_space(3)))

#if defined(__has_builtin)
#if __has_builtin(__builtin_amdgcn_global_load_async_to_lds_b128) && \
    __has_builtin(__builtin_amdgcn_s_wait_asynccnt)
#define USE_ASYNC 1
#endif
#endif
#ifndef USE_ASYNC
#define USE_ASYNC 0
#endif

namespace {
constexpr int   kNImg  = 256 * 256;    // 65536
constexpr int   kNSino = 360 * 512;    // 184320
constexpr int   kNnz   = 16777216;
constexpr float kCoef  = 0.01f;

constexpr int kThreads       = 256;    // 8 waves (wave32)
constexpr int kWavesPerBlock = kThreads / 32;
constexpr int kChunk         = 128;              // nnz per wave-chunk (32 lanes x 4)
constexpr int kNumChunks     = kNnz / kChunk;    // 131072
constexpr int kBlocks        = 1024;             // 8192 waves -> exactly 16 chunks/wave
constexpr int kDepth         = 4;                // async pipeline depth (ring buffer)
constexpr int kOutElems      = 4 * kNImg;        // 262144
}  // namespace

#if USE_ASYNC
typedef __attribute__((__vector_size__(16))) int v4i_t;
// global src (per-lane 16B) -> LDS dst (per-lane 16B), tracked by ASYNCcnt.
// cpol = 1 -> TH=NT: the nnz streams are read exactly once; keep them out of
// the L2 working set so pt + accumulator stay hot.
#define ASYNC_CP16(GSRC, LDST)                                                 \
  __builtin_amdgcn_global_load_async_to_lds_b128(                              \
      (AS1 v4i_t*)(unsigned long long)(const void*)(GSRC),                     \
      (AS3 v4i_t*)(unsigned int)(unsigned long long)(const void*)(LDST), 0, 1)
#endif

__device__ __forceinline__ void accumulate_one_pt(const float4* __restrict__ pt,
                                                  float* __restrict__ acc,
                                                  float val, int row, int col) {
  float4 q = pt[col];                      // single 16B L2-resident gather
  atomicAdd(acc + row,             val * q.x);
  atomicAdd(acc + row +     kNImg, val * q.y);
  atomicAdd(acc + row + 2 * kNImg, val * q.z);
  atomicAdd(acc + row + 3 * kNImg, val * q.w);
}

__device__ __forceinline__ void accumulate_one_p(const float* __restrict__ p,
                                                 float* __restrict__ acc,
                                                 float val, int row, int col) {
  float pa = p[col];
  float pb = p[col + kNSino];
  float pc = p[col + 2 * kNSino];
  float pd = p[col + 3 * kNSino];
  atomicAdd(acc + row,             val * pa);
  atomicAdd(acc + row +     kNImg, val * pb);
  atomicAdd(acc + row + 2 * kNImg, val * pc);
  atomicAdd(acc + row + 3 * kNImg, val * pd);
}

template <bool UsePt>
__global__ __launch_bounds__(kThreads) void spmm_scatter(
    const float* __restrict__ p, const float4* __restrict__ pt,
    const float* __restrict__ vals, const int* __restrict__ rows,
    const int* __restrict__ cols, float* __restrict__ acc) {
#if USE_ASYNC
  __shared__ __align__(16) float s_vals[kDepth][kWavesPerBlock][kChunk];
  __shared__ __align__(16) int   s_rows[kDepth][kWavesPerBlock][kChunk];
  __shared__ __align__(16) int   s_cols[kDepth][kWavesPerBlock][kChunk];

  const int wave  = threadIdx.x >> 5;
  const int lane  = threadIdx.x & 31;
  const int gwave = blockIdx.x * kWavesPerBlock + wave;
  const int W     = gridDim.x * kWavesPerBlock;

  if (gwave >= kNumChunks) return;
  const int iters = (kNumChunks - 1 - gwave) / W + 1;   // uniform (16) for our launch

  auto issue = [&](int it, int slot) {
    const int base = (gwave + it * W) * kChunk + lane * 4;  // 4 nnz / lane, 16B / stream
    ASYNC_CP16(vals + base, &s_vals[slot][wave][lane * 4]);
    ASYNC_CP16(rows + base, &s_rows[slot][wave][lane * 4]);
    ASYNC_CP16(cols + base, &s_cols[slot][wave][lane * 4]);
  };

  // Prologue: fill the pipeline (up to kDepth-1 chunks in flight).
  int issued = 0;
  for (; issued < kDepth - 1 && issued < iters; ++issued) issue(issued, issued & (kDepth - 1));

  for (int i = 0; i < iters; ++i) {
    if (issued < iters) {
      issue(issued, issued & (kDepth - 1));
      ++issued;
    }
    // Wait until chunk i's 3 async ops have landed; leave newer chunks in flight.
    const int ahead = issued - i - 1;   // chunks still in flight after this one
    if      (ahead >= 3) __builtin_amdgcn_s_wait_asynccnt(9);
    else if (ahead == 2) __builtin_amdgcn_s_wait_asynccnt(6);
    else if (ahead == 1) __builtin_amdgcn_s_wait_asynccnt(3);
    else                 __builtin_amdgcn_s_wait_asynccnt(0);
    asm volatile("" ::: "memory");      // order LDS reads after the wait

    const int slot = i & (kDepth - 1);
    const float4 v = *(const float4*)&s_vals[slot][wave][lane * 4];
    const int4   r = *(const int4*)&s_rows[slot][wave][lane * 4];
    const int4   q = *(const int4*)&s_cols[slot][wave][lane * 4];
    if (UsePt) {
      accumulate_one_pt(pt, acc, v.x, r.x, q.x);
      accumulate_one_pt(pt, acc, v.y, r.y, q.y);
      accumulate_one_pt(pt, acc, v.z, r.z, q.z);
      accumulate_one_pt(pt, acc, v.w, r.w, q.w);
    } else {
      accumulate_one_p(p, acc, v.x, r.x, q.x);
      accumulate_one_p(p, acc, v.y, r.y, q.y);
      accumulate_one_p(p, acc, v.z, r.z, q.z);
      accumulate_one_p(p, acc, v.w, r.w, q.w);
    }
  }
#else
  // Fallback: plain vectorized streaming loads + prefetch lookahead.
  const long long tid    = (long long)blockIdx.x * blockDim.x + threadIdx.x;
  const long long stride = (long long)gridDim.x * blockDim.x;
  const float4* v4 = (const float4*)vals;
  const int4*   r4 = (const int4*)rows;
  const int4*   c4 = (const int4*)cols;
  for (long long i = tid; i < kNnz / 4; i += stride) {
    if (i + stride < kNnz / 4) {
      __builtin_prefetch(&v4[i + stride], 0, 0);
      __builtin_prefetch(&r4[i + stride], 0, 0);
      __builtin_prefetch(&c4[i + stride], 0, 0);
    }
    const float4 v = v4[i];
    const int4   r = r4[i];
    const int4   q = c4[i];
    if (UsePt) {
      accumulate_one_pt(pt, acc, v.x, r.x, q.x);
      accumulate_one_pt(pt, acc, v.y, r.y, q.y);
      accumulate_one_pt(pt, acc, v.z, r.z, q.z);
      accumulate_one_pt(pt, acc, v.w, r.w, q.w);
    } else {
      accumulate_one_p(p, acc, v.x, r.x, q.x);
      accumulate_one_p(p, acc, v.y, r.y, q.y);
      accumulate_one_p(p, acc, v.z, r.z, q.z);
      accumulate_one_p(p, acc, v.w, r.w, q.w);
    }
  }
#endif
}

// pt[col] = (p[0,col], p[1,col], p[2,col], p[3,col]) : one float4 gather per nnz later.
__global__ void transpose_p(const float* __restrict__ p, float4* __restrict__ pt) {
  const int i = blockIdx.x * blockDim.x + threadIdx.x;
  if (i < kNSino) {
    pt[i] = make_float4(p[i], p[i + kNSino], p[i + 2 * kNSino], p[i + 3 * kNSino]);
  }
}

__global__ void zero_out(float* __restrict__ out) {
  const int i = blockIdx.x * blockDim.x + threadIdx.x;
  if (i < kOutElems) out[i] = 0.0f;
}

__global__ void finalize(float* __restrict__ out, const float* __restrict__ bias) {
  const int i = blockIdx.x * blockDim.x + threadIdx.x;
  if (i < kOutElems) out[i] = out[i] * kCoef + bias[i & (kNImg - 1)];
}

extern "C" void kernel_launch(void* const* d_in, const int* in_sizes, int n_in,
                              void* d_out, int out_size, void* d_ws, size_t ws_size,
                              hipStream_t stream) {
  (void)in_sizes; (void)n_in; (void)out_size;
  const float* p    = (const float*)d_in[0];
  const float* vals = (const float*)d_in[1];
  const int*   rows = (const int*)d_in[2];
  const int*   cols = (const int*)d_in[3];
  const float* bias = (const float*)d_in[4];
  float* out = (float*)d_out;

  const int outBlocks = (kOutElems + kThreads - 1) / kThreads;
  zero_out<<<outBlocks, kThreads, 0, stream>>>(out);

  const bool usePt = ws_size >= (size_t)kNSino * sizeof(float4);
  if (usePt) {
    float4* pt = (float4*)d_ws;
    transpose_p<<<(kNSino + kThreads - 1) / kThreads, kThreads, 0, stream>>>(p, pt);
    spmm_scatter<true><<<kBlocks, kThreads, 0, stream>>>(p, pt, vals, rows, cols, out);
  } else {
    spmm_scatter<false><<<kBlocks, kThreads, 0, stream>>>(p, nullptr, vals, rows, cols, out);
  }

  finalize<<<outBlocks, kThreads, 0, stream>>>(out, bias);
}